// Decompose_85727547228274
// MI455X (gfx1250) — compile-verified
//
#include <hip/hip_runtime.h>

typedef float v2f __attribute__((ext_vector_type(2)));
typedef float v8f __attribute__((ext_vector_type(8)));

#define B_DIM 32
#define C_DIM 512
#define L_DIM 2048

// One wave32 per (channel, batch-half). The wave scans L in 16-wide chunks:
//   D(16 batches x 16 positions) = X^T (16x16) * M^T (16x16)  via 4x WMMA f32 16x16x4
// then adds the carry term (1-a)^{pos+1} * carry and broadcasts the pos=15
// column as the next chunk's carry with ds_swizzle.
__global__ __launch_bounds__(32) void ema_scan_wmma(const float* __restrict__ x,
                                                    const float* __restrict__ alpha,
                                                    float* __restrict__ out) {
    const int lane = threadIdx.x & 31;
    const int wid  = blockIdx.x;          // 0..1023
    const int c    = wid >> 1;            // channel 0..511
    const int b0   = (wid & 1) * 16;      // batch group base (0 or 16)

    const float a   = alpha[c];
    const float omp = 1.0f - a;

    const int pos = lane & 15;            // N index: position within chunk
    const int hi  = lane >> 4;            // lane half (selects K / batch half)

    // ---- B operand: M^T, triangular coefficients, 4 K-groups x 2 VGPRs ----
    // B[j][pos] = a * (1-a)^(pos-j) for j <= pos, else 0.
    // Layout (4x16 per WMMA): v0 holds K = 4*kg + 2*hi, v1 holds K = that+1; N = pos.
    v2f bmat[4];
    #pragma unroll
    for (int kg = 0; kg < 4; ++kg) {
        #pragma unroll
        for (int r = 0; r < 2; ++r) {
            const int j = 4 * kg + 2 * hi + r;
            const int e = pos - j;
            float v = a;
            #pragma unroll
            for (int t = 0; t < 15; ++t) v = (t < e) ? v * omp : v;   // v = a*(1-a)^e
            bmat[kg][r] = (e >= 0) ? v : 0.0f;
        }
    }

    // Per-lane decay for the incoming carry: (1-a)^(pos+1)
    float dpow = omp;
    #pragma unroll
    for (int t = 0; t < 15; ++t) dpow = (t < pos) ? dpow * omp : dpow;

    // ---- Addressing ----
    // A operand (X^T, 16x4 per WMMA): lane M = batch = pos-index of lane,
    // K striped: v0 -> K = 4*kg + 2*hi, v1 -> +1  => one b64 load per K-group.
    const float* ap = x + ((size_t)(b0 + (lane & 15)) * C_DIM + c) * (size_t)L_DIM + 2 * hi;

    // D/C layout: VGPR g, lane half hi -> batch = b0 + g + 8*hi, N = pos.
    float* op[8];
    float  carry[8];
    #pragma unroll
    for (int g = 0; g < 8; ++g) {
        const size_t bofs = ((size_t)(b0 + g + 8 * hi) * C_DIM + c) * (size_t)L_DIM;
        op[g]    = out + bofs + pos;
        carry[g] = x[bofs];               // seed carry with x[b, c, 0] (makes y0 == x0 exact)
    }

    for (int l0 = 0; l0 < L_DIM; l0 += 16) {
        // Load this chunk's A operands (4 x b64 per lane).
        v2f a0 = *(const v2f*)(ap + l0 + 0);
        v2f a1 = *(const v2f*)(ap + l0 + 4);
        v2f a2 = *(const v2f*)(ap + l0 + 8);
        v2f a3 = *(const v2f*)(ap + l0 + 12);

        v8f d = {};
        d = __builtin_amdgcn_wmma_f32_16x16x4_f32(false, a0, false, bmat[0], (short)0, d, false, false);
        d = __builtin_amdgcn_wmma_f32_16x16x4_f32(false, a1, false, bmat[1], (short)0, d, false, false);
        d = __builtin_amdgcn_wmma_f32_16x16x4_f32(false, a2, false, bmat[2], (short)0, d, false, false);
        d = __builtin_amdgcn_wmma_f32_16x16x4_f32(false, a3, false, bmat[3], (short)0, d, false, false);

        #pragma unroll
        for (int g = 0; g < 8; ++g) {
            const float y = d[g] + dpow * carry[g];
            // Broadcast pos=15 value of each batch half to its whole half:
            // src_lane = (lane & 0x10) | 0x0F  -> 15 for lanes 0-15, 31 for lanes 16-31.
            const int yi = __builtin_amdgcn_ds_swizzle(__builtin_bit_cast(int, y), 0x1F0);
            carry[g] = __builtin_bit_cast(float, yi);
            op[g][l0] = y;                // coalesced: 16 contiguous floats per lane half
        }
    }
}

extern "C" void kernel_launch(void* const* d_in, const int* in_sizes, int n_in,
                              void* d_out, int out_size, void* d_ws, size_t ws_size,
                              hipStream_t stream) {
    (void)in_sizes; (void)n_in; (void)d_ws; (void)ws_size; (void)out_size;
    const float* x     = (const float*)d_in[0];
    const float* alpha = (const float*)d_in[1];
    float* out         = (float*)d_out;

    // 512 channels * 2 batch-halves = 1024 independent single-wave workgroups.
    ema_scan_wmma<<<dim3(C_DIM * 2), dim3(32), 0, stream>>>(x, alpha, out);
}